// FwdIntegration_16501264351750
// MI455X (gfx1250) — compile-verified
//
#include <hip/hip_runtime.h>
#include <math.h>

// ---------------------------------------------------------------------------
// FLASH EPDiff forward integration on 96^3x3, 10 Euler steps.
// 3D FFTs run as 96x96 complex DFT GEMMs on the CDNA5 fp32 WMMA pipe
// (v_wmma_f32_16x16x4_f32). Axis X/Y passes read strided-coalesced straight
// from L2; axis Z (contraction dim contiguous) is staged through LDS with
// padded panels for conflict-free ds_load_b64. Working set ~75MB = L2-resident.
// ---------------------------------------------------------------------------

#define NX    96
#define NPLN  (96 * 96)        // 9216
#define NGRID (96 * 96 * 96)   // 884736
#define NFLD  (NGRID * 3)      // 2654208
#define DT_C  0.1f
#define CN_A  0.0025f
#define PI_F  3.14159265358979323846f

typedef __attribute__((ext_vector_type(2))) float v2f;
typedef __attribute__((ext_vector_type(8))) float v8f;

// fp32 WMMA: D(16x16,f32) = A(16x4,f32) * B(4x16,f32) + C
__device__ __forceinline__ v8f wmma4(v2f a, v2f b, v8f c) {
  return __builtin_amdgcn_wmma_f32_16x16x4_f32(
      /*neg_a=*/false, a, /*neg_b=*/false, b,
      /*c_mod=*/(short)0, c, /*reuse_a=*/false, /*reuse_b=*/false);
}

// ---------------------------------------------------------------------------
// Generic strided-batch complex DFT GEMM (used for axes X and Y, where the
// j (lane) dimension is memory-contiguous -> coalesced global_load_b64).
//   out[b, i, j] = sum_k W[i,k] * in[b, k, j]
//   element address = b*Sb + (k or i)*Sk + j*Sj    (float2-complex elements)
// Lane layout (CDNA5 ISA 7.12.2, 32-bit matrices, wave32):
//   A 16x4 : M = lane%16 ; reg e -> K = (lane/16)*2 + e
//   B 4x16 : N = lane%16 ; reg e -> K = e*2 + lane/16
//   C/D    : N = lane%16 ; reg r -> M = r + (lane/16)*8
// ---------------------------------------------------------------------------
__global__ void dft_axis_kernel(const float2* __restrict__ in,
                                float2* __restrict__ out,
                                const float* __restrict__ Wr,
                                const float* __restrict__ Wi,
                                int B, int J, int Sb, int Sk, int Sj,
                                int tiles) {
  int wave = blockIdx.x * (blockDim.x >> 5) + (threadIdx.x >> 5);
  if (wave >= tiles) return;                 // wave-uniform exit
  int jtc = J >> 4;
  int jt  = wave % jtc;
  int it  = (wave / jtc) % 6;                // 96/16 = 6 row tiles
  int b   = wave / (jtc * 6);

  int lane = threadIdx.x & 31;
  int half = lane >> 4;
  int lm   = lane & 15;
  int i0 = it << 4, j0 = jt << 4;

  const float* wr = Wr + (i0 + lm) * NX;
  const float* wi = Wi + (i0 + lm) * NX;
  int inBase = b * Sb + (j0 + lm) * Sj;

  v8f arr = {0.f,0.f,0.f,0.f,0.f,0.f,0.f,0.f};
  v8f aii = arr, ari = arr, air = arr;

  for (int k = 0; k < NX; k += 4) {
    int ka = k + (half << 1);
    v2f a_r = { wr[ka], wr[ka + 1] };
    v2f a_i = { wi[ka], wi[ka + 1] };
    float2 x0 = in[inBase + (k + half) * Sk];
    float2 x1 = in[inBase + (k + 2 + half) * Sk];
    v2f b_r = { x0.x, x1.x };
    v2f b_i = { x0.y, x1.y };
    arr = wmma4(a_r, b_r, arr);              // Re*Re
    aii = wmma4(a_i, b_i, aii);              // Im*Im
    ari = wmma4(a_r, b_i, ari);              // Re*Im
    air = wmma4(a_i, b_r, air);              // Im*Re
  }

  int outBase = b * Sb + (j0 + lm) * Sj;
#pragma unroll
  for (int r = 0; r < 8; ++r) {
    int i = i0 + (half << 3) + r;
    out[outBase + i * Sk] = make_float2(arr[r] - aii[r], ari[r] + air[r]);
  }
}

// ---------------------------------------------------------------------------
// Axis-Z DFT GEMM, LDS-staged. Memory layout along Z: addr = b*9216 + j*96 + k
// (k = z contiguous). A naive WMMA B-fragment load here would stride lanes by
// 768B (32 cache lines per load). Instead: one block = (b, j-tile); 6 waves.
//  1) 192 threads load the 16(j) x 96(k) complex panel coalesced into LDS
//     (rows padded to 97 float2 -> lane reads hit distinct bank pairs)
//  2) each wave computes one 16(i) x 16(j) tile from LDS fragments
//  3) tiles staged to an output LDS panel, stored coalesced along i
// ---------------------------------------------------------------------------
#define ZPAD 97
__global__ void dft_axis_z_kernel(const float2* __restrict__ in,
                                  float2* __restrict__ out,
                                  const float* __restrict__ Wr,
                                  const float* __restrict__ Wi) {
  __shared__ float2 pin [16 * ZPAD];
  __shared__ float2 pout[16 * ZPAD];

  int b  = blockIdx.x / 6;                   // x-slab
  int jt = blockIdx.x - b * 6;               // y tile
  int j0 = jt << 4;
  int tid = threadIdx.x;                     // 0..191 (6 waves)

  const float2* src = in + b * NPLN;
#pragma unroll
  for (int r = 0; r < 8; ++r) {              // 1536 float2, coalesced in k
    int idx = tid + 192 * r;
    int j = idx / NX, k = idx - j * NX;
    pin[j * ZPAD + k] = src[(j0 + j) * NX + k];
  }
  __syncthreads();

  int wave = tid >> 5;                       // i-tile 0..5
  int lane = tid & 31;
  int half = lane >> 4, lm = lane & 15;
  int i0 = wave << 4;

  const float* wr = Wr + (i0 + lm) * NX;
  const float* wi = Wi + (i0 + lm) * NX;
  const float2* bp = pin + lm * ZPAD;

  v8f arr = {0.f,0.f,0.f,0.f,0.f,0.f,0.f,0.f};
  v8f aii = arr, ari = arr, air = arr;

  for (int k = 0; k < NX; k += 4) {
    int ka = k + (half << 1);
    v2f a_r = { wr[ka], wr[ka + 1] };
    v2f a_i = { wi[ka], wi[ka + 1] };
    float2 x0 = bp[k + half];                // ds_load_b64, conflict-free
    float2 x1 = bp[k + 2 + half];
    v2f b_r = { x0.x, x1.x };
    v2f b_i = { x0.y, x1.y };
    arr = wmma4(a_r, b_r, arr);
    aii = wmma4(a_i, b_i, aii);
    ari = wmma4(a_r, b_i, ari);
    air = wmma4(a_i, b_r, air);
  }

  float2* op = pout + lm * ZPAD;
#pragma unroll
  for (int r = 0; r < 8; ++r) {
    int i = i0 + (half << 3) + r;
    op[i] = make_float2(arr[r] - aii[r], ari[r] + air[r]);
  }
  __syncthreads();

  float2* dst = out + b * NPLN;
#pragma unroll
  for (int r = 0; r < 8; ++r) {              // store coalesced along i
    int idx = tid + 192 * r;
    int j = idx / NX, i = idx - j * NX;
    dst[(j0 + j) * NX + i] = pout[j * ZPAD + i];
  }
}

// ---------------------------------------------------------------------------
// Precompute kernels (on-stream each call; deterministic, capture-safe)
// ---------------------------------------------------------------------------
__global__ void build_dft_kernel(float* Wfr, float* Wfi, float* Wbr, float* Wbi) {
  int id = blockIdx.x * blockDim.x + threadIdx.x;
  if (id >= NX * NX) return;
  int i = id / NX, k = id % NX;
  int ph = (i * k) % NX;                       // exact phase reduction
  float a = -2.0f * PI_F * (float)ph / (float)NX;
  Wfr[id] = cosf(a);                           // forward: exp(-2*pi*i*jk/N)
  Wfi[id] = sinf(a);
  Wbr[id] = cosf(a)  * (1.0f / (float)NX);     // inverse: exp(+..)/N per axis
  Wbi[id] = -sinf(a) * (1.0f / (float)NX);
}

__global__ void build_spec_kernel(float* Lsp, float* Ksp) {
  int i = blockIdx.x * blockDim.x + threadIdx.x;
  if (i >= NGRID) return;
  int z = i % NX, y = (i / NX) % NX, x = i / NPLN;
  const float c = 2.0f * PI_F / (float)NX;
  float lap = 2.0f * (3.0f - cosf(c * (float)x) - cosf(c * (float)y)
                           - cosf(c * (float)z));
  float l = 1.0f + CN_A * lap;   // sigma=1, cn_gamma=1, s=1, gamma=0
  Lsp[i] = l;
  Ksp[i] = 1.0f / l;
}

// ---------------------------------------------------------------------------
// Pointwise / stencil kernels (L2-resident traffic)
// ---------------------------------------------------------------------------
__global__ void zero_kernel(float* p, int n) {
  int i = blockIdx.x * blockDim.x + threadIdx.x;
  if (i < n) p[i] = 0.0f;
}
__global__ void copy_kernel(float* dst, const float* src, int n) {
  int i = blockIdx.x * blockDim.x + threadIdx.x;
  if (i < n) dst[i] = src[i];
}
__global__ void pack_kernel(const float* __restrict__ f, int c, float2* __restrict__ o) {
  int i = blockIdx.x * blockDim.x + threadIdx.x;
  if (i < NGRID) o[i] = make_float2(f[i * 3 + c], 0.0f);
}
// mode 0: f[.,c] = Re(in)          (m = L v)
// mode 1: f[.,c] -= DT * Re(in)    (Euler v-update fused into iFFT epilogue)
__global__ void unpack_kernel(const float2* __restrict__ in, float* __restrict__ f,
                              int c, int mode) {
  int i = blockIdx.x * blockDim.x + threadIdx.x;
  if (i >= NGRID) return;
  float r = in[i].x;
  f[i * 3 + c] = (mode == 0) ? r : (f[i * 3 + c] - DT_C * r);
}
__global__ void specmul_kernel(float2* __restrict__ a, const float* __restrict__ s) {
  int i = blockIdx.x * blockDim.x + threadIdx.x;
  if (i >= NGRID) return;
  float2 v = a[i]; float ss = s[i];
  a[i] = make_float2(v.x * ss, v.y * ss);
}

__device__ __forceinline__ void nbr6(int i, int& xp, int& xm, int& yp, int& ym,
                                     int& zp, int& zm) {
  int z = i % NX, y = (i / NX) % NX, x = i / NPLN;
  xp = (x == NX - 1) ? i - (NX - 1) * NPLN : i + NPLN;
  xm = (x == 0)      ? i + (NX - 1) * NPLN : i - NPLN;
  yp = (y == NX - 1) ? i - (NX - 1) * NX   : i + NX;
  ym = (y == 0)      ? i + (NX - 1) * NX   : i - NX;
  zp = (z == NX - 1) ? i - (NX - 1)        : i + 1;
  zm = (z == 0)      ? i + (NX - 1)        : i - 1;
}

// dis_{n+1} = dis_n - DT * (Jac(dis_n) . v + v)   (periodic central diffs)
__global__ void step_dis_kernel(const float* __restrict__ dis,
                                const float* __restrict__ v,
                                float* __restrict__ dout) {
  int i = blockIdx.x * blockDim.x + threadIdx.x;
  if (i >= NGRID) return;
  int xp, xm, yp, ym, zp, zm; nbr6(i, xp, xm, yp, ym, zp, zm);
  float vx = v[i * 3 + 0], vy = v[i * 3 + 1], vz = v[i * 3 + 2];
#pragma unroll
  for (int c = 0; c < 3; ++c) {
    float gx = 0.5f * (dis[xp * 3 + c] - dis[xm * 3 + c]);
    float gy = 0.5f * (dis[yp * 3 + c] - dis[ym * 3 + c]);
    float gz = 0.5f * (dis[zp * 3 + c] - dis[zm * 3 + c]);
    float w = gx * vx + gy * vy + gz * vz;
    dout[i * 3 + c] = dis[i * 3 + c] - DT_C * (w + v[i * 3 + c]);
  }
}

// t_i = sum_c d_i(v)_c * m_c  +  sum_j d_j(m_i * v_j)
__global__ void adterms_kernel(const float* __restrict__ v,
                               const float* __restrict__ m,
                               float* __restrict__ t) {
  int i = blockIdx.x * blockDim.x + threadIdx.x;
  if (i >= NGRID) return;
  int xp, xm, yp, ym, zp, zm; nbr6(i, xp, xm, yp, ym, zp, zm);
  float mcs[3] = {m[i * 3 + 0], m[i * 3 + 1], m[i * 3 + 2]};
  float t1[3] = {0.f, 0.f, 0.f};
#pragma unroll
  for (int c = 0; c < 3; ++c) {
    t1[0] += 0.5f * (v[xp * 3 + c] - v[xm * 3 + c]) * mcs[c];
    t1[1] += 0.5f * (v[yp * 3 + c] - v[ym * 3 + c]) * mcs[c];
    t1[2] += 0.5f * (v[zp * 3 + c] - v[zm * 3 + c]) * mcs[c];
  }
#pragma unroll
  for (int d = 0; d < 3; ++d) {
    float t2 = 0.5f * (m[xp * 3 + d] * v[xp * 3 + 0] - m[xm * 3 + d] * v[xm * 3 + 0]
                     + m[yp * 3 + d] * v[yp * 3 + 1] - m[ym * 3 + d] * v[ym * 3 + 1]
                     + m[zp * 3 + d] * v[zp * 3 + 2] - m[zm * 3 + d] * v[zm * 3 + 2]);
    t[i * 3 + d] = t1[d] + t2;
  }
}

// ---------------------------------------------------------------------------
// Host orchestration
// ---------------------------------------------------------------------------
static inline void launch_dft(const float2* in, float2* out, const float* Wr,
                              const float* Wi, int B, int J, int Sb, int Sk,
                              int Sj, hipStream_t s) {
  int tiles = 6 * (J / 16) * B;              // 3456 for X and Y passes
  int blocks = (tiles + 7) / 8;              // 8 waves (one tile each) / block
  dft_axis_kernel<<<blocks, 256, 0, s>>>(in, out, Wr, Wi, B, J, Sb, Sk, Sj, tiles);
}

// out = ifft3( spec * fft3(channel c of fin) ).real, per channel; mode 1 fuses
// the Euler update  fout[.,c] -= DT * result  into the epilogue.
static void apply_op(const float* fin, float* fout, int mode, const float* spec,
                     float2* C0, float2* C1, const float* Wfr, const float* Wfi,
                     const float* Wbr, const float* Wbi, hipStream_t s) {
  const int PB = NGRID / 256, PT = 256;      // 3456 blocks
  for (int c = 0; c < 3; ++c) {
    pack_kernel<<<PB, PT, 0, s>>>(fin, c, C0);
    launch_dft(C0, C1, Wfr, Wfi, 1,  NPLN, 0,    NPLN, 1,  s);   // fwd X
    launch_dft(C1, C0, Wfr, Wfi, NX, NX,   NPLN, NX,   1,  s);   // fwd Y
    dft_axis_z_kernel<<<NX * 6, 192, 0, s>>>(C0, C1, Wfr, Wfi);  // fwd Z (LDS)
    specmul_kernel<<<PB, PT, 0, s>>>(C1, spec);
    dft_axis_z_kernel<<<NX * 6, 192, 0, s>>>(C1, C0, Wbr, Wbi);  // inv Z (LDS)
    launch_dft(C0, C1, Wbr, Wbi, NX, NX,   NPLN, NX,   1,  s);   // inv Y
    launch_dft(C1, C0, Wbr, Wbi, 1,  NPLN, 0,    NPLN, 1,  s);   // inv X
    unpack_kernel<<<PB, PT, 0, s>>>(C0, fout, c, mode);
  }
}

extern "C" void kernel_launch(void* const* d_in, const int* in_sizes, int n_in,
                              void* d_out, int out_size, void* d_ws, size_t ws_size,
                              hipStream_t stream) {
  const float* v0 = (const float*)d_in[0];
  float* out = (float*)d_out;

  // Workspace carve-up (~74.5 MB total; fully L2-resident on MI455X)
  float* p    = (float*)d_ws;
  float* disA = p; p += NFLD;
  float* disB = p; p += NFLD;
  float* v    = p; p += NFLD;
  float* m    = p; p += NFLD;
  float* t    = p; p += NFLD;
  float2* C0  = (float2*)p; p += 2 * NGRID;
  float2* C1  = (float2*)p; p += 2 * NGRID;
  float* Wfr  = p; p += NX * NX;
  float* Wfi  = p; p += NX * NX;
  float* Wbr  = p; p += NX * NX;
  float* Wbi  = p; p += NX * NX;
  float* Lsp  = p; p += NGRID;
  float* Ksp  = p; p += NGRID;

  const int PB = NGRID / 256, PT = 256;
  build_dft_kernel<<<(NX * NX + 255) / 256, 256, 0, stream>>>(Wfr, Wfi, Wbr, Wbi);
  build_spec_kernel<<<PB, PT, 0, stream>>>(Lsp, Ksp);
  zero_kernel<<<(NFLD + 255) / 256, 256, 0, stream>>>(disA, NFLD);
  hipMemcpyAsync(v, v0, (size_t)NFLD * sizeof(float),
                 hipMemcpyDeviceToDevice, stream);

  float* dcur = disA; float* dnxt = disB;
  for (int step = 0; step < 10; ++step) {
    // dis update uses old dis and old v (reference order)
    step_dis_kernel<<<PB, PT, 0, stream>>>(dcur, v, dnxt);
    // m = L v   (FFT -> *L -> iFFT, per channel, WMMA DFT GEMMs)
    apply_op(v, m, 0, Lsp, C0, C1, Wfr, Wfi, Wbr, Wbi, stream);
    // t = t1 + t2 from v, m
    adterms_kernel<<<PB, PT, 0, stream>>>(v, m, t);
    // v -= DT * K(t)   (update fused into inverse-FFT epilogue)
    apply_op(t, v, 1, Ksp, C0, C1, Wfr, Wfi, Wbr, Wbi, stream);
    float* tmp = dcur; dcur = dnxt; dnxt = tmp;
  }
  copy_kernel<<<(NFLD + 255) / 256, 256, 0, stream>>>(out, dcur, NFLD);
}